// Criterian_57552561766425
// MI455X (gfx1250) — compile-verified
//
#include <hip/hip_runtime.h>
#include <hip/hip_bf16.h>

// Problem constants (B=64, C=2, H=W=512)
#define NTOT 16777216u   // 64*512*512 elements per channel
#define HWC  262144u     // 512*512 (one plane)
#define KMAX 1000u
#define NB0  2048        // level-0 bins: float bits [30:20]
#define NB1  1024        // level-1/2 bins: 10 bits each

typedef __attribute__((ext_vector_type(2))) float v2f;
typedef __attribute__((ext_vector_type(4))) float v4f;
typedef __attribute__((ext_vector_type(8))) float v8f;

struct OhnmWs {
    unsigned int h0_cnt[2][NB0];
    float        h0_sum[2][NB0];
    unsigned int h1_cnt[2][NB1];
    float        h1_sum[2][NB1];
    unsigned int h2_cnt[2][NB1];
    float        h2_sum[2][NB1];
    unsigned int num_pos[2];
    float        pos_sum[2];
    unsigned int k[2];
    unsigned int k_rem[2];
    unsigned int prefix[2];
    unsigned int cnt_gt[2];
    float        sum_gt[2];
};

// Wave32 sum reduction on the matrix pipe: D = A x ones + 0.
// Each lane contributes x at one (M,K) slot of the 16x4 A tile; with B = all
// ones, D[m][n] = rowsum_m. Lanes 0-15 hold rows 0-7 in their 8 D VGPRs,
// lanes 16-31 hold rows 8-15, so (sum of my 8) + shfl_xor(...,16) = wave sum.
__device__ __forceinline__ float wave_reduce_f32(float x) {
#if defined(__HIP_DEVICE_COMPILE__) && __has_builtin(__builtin_amdgcn_wmma_f32_16x16x4_f32)
    v2f a; a[0] = x;    a[1] = 0.0f;
    v2f o; o[0] = 1.0f; o[1] = 1.0f;
    v8f c = {};
    c = __builtin_amdgcn_wmma_f32_16x16x4_f32(false, a, false, o,
                                              (short)0, c, false, false);
    float s = ((c[0] + c[1]) + (c[2] + c[3])) + ((c[4] + c[5]) + (c[6] + c[7]));
    s += __shfl_xor(s, 16, 32);
    return s;
#else
    for (int off = 16; off > 0; off >>= 1) x += __shfl_xor(x, off, 32);
    return x;
#endif
}

__global__ void ohnm_zero_ws(unsigned int* __restrict__ p, unsigned int nwords) {
    unsigned int i = blockIdx.x * blockDim.x + threadIdx.x;
    if (i < nwords) p[i] = 0u;
}

// Pass 1: losses, positive stats, level-0 histogram (count + sum) per channel.
// Weight streams are single-use -> non-temporal loads so they do not evict the
// output/map working set (256 MiB, mostly fits the 192 MB L2) that passes 2/3
// re-read.
__global__ void __launch_bounds__(256) ohnm_pass1(
    const float* __restrict__ out,
    const float* __restrict__ cmap, const float* __restrict__ amap,
    const float* __restrict__ cw,   const float* __restrict__ aw,
    OhnmWs* __restrict__ ws)
{
    __shared__ unsigned int s_cnt[2][NB0];
    __shared__ float        s_sum[2][NB0];
    __shared__ unsigned int s_npos[2];
    __shared__ float        s_psum[2];

    for (unsigned int i = threadIdx.x; i < NB0; i += blockDim.x) {
        s_cnt[0][i] = 0u;  s_cnt[1][i] = 0u;
        s_sum[0][i] = 0.f; s_sum[1][i] = 0.f;
    }
    if (threadIdx.x < 2u) { s_npos[threadIdx.x] = 0u; s_psum[threadIdx.x] = 0.f; }
    __syncthreads();

    unsigned int npos0 = 0u, npos1 = 0u;
    float psum0 = 0.f, psum1 = 0.f;

    const unsigned int nvec   = NTOT / 4u;
    const unsigned int stride = gridDim.x * blockDim.x;
    for (unsigned int v = blockIdx.x * blockDim.x + threadIdx.x; v < nvec; v += stride) {
        const unsigned int idx = v * 4u;
        const unsigned int b   = idx >> 18;          // / HWC
        const unsigned int hw  = idx & (HWC - 1u);

        // Prefetch next grid-stride tile (global_prefetch_b8).
        const unsigned int vn = v + stride;
        if (vn < nvec) {
            const unsigned int idn = vn * 4u;
            const unsigned int bn  = idn >> 18;
            const unsigned int hwn = idn & (HWC - 1u);
            __builtin_prefetch(out + (size_t)(bn * 2u) * HWC + hwn, 0, 3);
            __builtin_prefetch(out + (size_t)(bn * 2u + 1u) * HWC + hwn, 0, 3);
            __builtin_prefetch(cmap + idn, 0, 3);
            __builtin_prefetch(amap + idn, 0, 3);
            __builtin_prefetch(cw + idn, 0, 0);   // streaming, no reuse
            __builtin_prefetch(aw + idn, 0, 0);   // streaming, no reuse
        }

        const float4 pc = *(const float4*)(out + (size_t)(b * 2u) * HWC + hw);
        const float4 pa = *(const float4*)(out + (size_t)(b * 2u + 1u) * HWC + hw);
        const float4 tc = *(const float4*)(cmap + idx);
        const float4 ta = *(const float4*)(amap + idx);
        const v4f    wc = __builtin_nontemporal_load((const v4f*)(cw + idx));
        const v4f    wa = __builtin_nontemporal_load((const v4f*)(aw + idx));

#pragma unroll
        for (int j = 0; j < 4; ++j) {
            float p0 = (&pc.x)[j], t0 = (&tc.x)[j], w0 = wc[j];
            float d0 = p0 - t0;
            float l0 = d0 * d0;
            if (t0 != 0.f) { npos0++; psum0 += l0 * w0; }
            else {
                unsigned int bin = __float_as_uint(l0) >> 20;
                atomicAdd(&s_cnt[0][bin], 1u);
                atomicAdd(&s_sum[0][bin], l0);
            }
            float p1 = (&pa.x)[j], t1 = (&ta.x)[j], w1 = wa[j];
            float d1 = p1 - t1;
            float l1 = d1 * d1;
            if (t1 != 0.f) { npos1++; psum1 += l1 * w1; }
            else {
                unsigned int bin = __float_as_uint(l1) >> 20;
                atomicAdd(&s_cnt[1][bin], 1u);
                atomicAdd(&s_sum[1][bin], l1);
            }
        }
    }

    // Wave-level float reduction on the matrix pipe, then block combine.
    float r0 = wave_reduce_f32(psum0);
    float r1 = wave_reduce_f32(psum1);
    if ((threadIdx.x & 31u) == 0u) {
        atomicAdd(&s_psum[0], r0);
        atomicAdd(&s_psum[1], r1);
    }
    atomicAdd(&s_npos[0], npos0);
    atomicAdd(&s_npos[1], npos1);
    __syncthreads();

    for (unsigned int i = threadIdx.x; i < NB0; i += blockDim.x) {
        if (s_cnt[0][i])        atomicAdd(&ws->h0_cnt[0][i], s_cnt[0][i]);
        if (s_cnt[1][i])        atomicAdd(&ws->h0_cnt[1][i], s_cnt[1][i]);
        if (s_sum[0][i] != 0.f) atomicAdd(&ws->h0_sum[0][i], s_sum[0][i]);
        if (s_sum[1][i] != 0.f) atomicAdd(&ws->h0_sum[1][i], s_sum[1][i]);
    }
    if (threadIdx.x == 0u) {
        atomicAdd(&ws->num_pos[0], s_npos[0]);
        atomicAdd(&ws->num_pos[1], s_npos[1]);
        atomicAdd(&ws->pos_sum[0], s_psum[0]);
        atomicAdd(&ws->pos_sum[1], s_psum[1]);
    }
}

// Radix-refine passes (level 1: bits[19:10] within the level-0 bin;
// level 2: bits[9:0] within the 21-bit prefix). Regular temporal loads so the
// L2-resident output/map lines from pass 1 are hit.
__global__ void __launch_bounds__(256) ohnm_refine(
    const float* __restrict__ out,
    const float* __restrict__ cmap, const float* __restrict__ amap,
    OhnmWs* __restrict__ ws, int level)
{
    __shared__ unsigned int s_cnt[2][NB1];
    __shared__ float        s_sum[2][NB1];

    for (unsigned int i = threadIdx.x; i < NB1; i += blockDim.x) {
        s_cnt[0][i] = 0u;  s_cnt[1][i] = 0u;
        s_sum[0][i] = 0.f; s_sum[1][i] = 0.f;
    }
    __syncthreads();

    const unsigned int pref0 = ws->prefix[0];
    const unsigned int pref1 = ws->prefix[1];

    const unsigned int nvec   = NTOT / 4u;
    const unsigned int stride = gridDim.x * blockDim.x;
    for (unsigned int v = blockIdx.x * blockDim.x + threadIdx.x; v < nvec; v += stride) {
        const unsigned int idx = v * 4u;
        const unsigned int b   = idx >> 18;
        const unsigned int hw  = idx & (HWC - 1u);

        const unsigned int vn = v + stride;
        if (vn < nvec) {
            const unsigned int idn = vn * 4u;
            const unsigned int bn  = idn >> 18;
            const unsigned int hwn = idn & (HWC - 1u);
            __builtin_prefetch(out + (size_t)(bn * 2u) * HWC + hwn, 0, 3);
            __builtin_prefetch(out + (size_t)(bn * 2u + 1u) * HWC + hwn, 0, 3);
            __builtin_prefetch(cmap + idn, 0, 3);
            __builtin_prefetch(amap + idn, 0, 3);
        }

        const float4 pc = *(const float4*)(out + (size_t)(b * 2u) * HWC + hw);
        const float4 pa = *(const float4*)(out + (size_t)(b * 2u + 1u) * HWC + hw);
        const float4 tc = *(const float4*)(cmap + idx);
        const float4 ta = *(const float4*)(amap + idx);

#pragma unroll
        for (int j = 0; j < 4; ++j) {
            float t0 = (&tc.x)[j];
            if (t0 == 0.f) {
                float d0 = (&pc.x)[j] - t0;
                unsigned int u = __float_as_uint(d0 * d0);
                if (level == 1) {
                    if ((u >> 20) == pref0) {
                        unsigned int bin = (u >> 10) & (NB1 - 1u);
                        atomicAdd(&s_cnt[0][bin], 1u);
                        atomicAdd(&s_sum[0][bin], __uint_as_float(u));
                    }
                } else {
                    if ((u >> 10) == pref0) {
                        unsigned int bin = u & (NB1 - 1u);
                        atomicAdd(&s_cnt[0][bin], 1u);
                        atomicAdd(&s_sum[0][bin], __uint_as_float(u));
                    }
                }
            }
            float t1 = (&ta.x)[j];
            if (t1 == 0.f) {
                float d1 = (&pa.x)[j] - t1;
                unsigned int u = __float_as_uint(d1 * d1);
                if (level == 1) {
                    if ((u >> 20) == pref1) {
                        unsigned int bin = (u >> 10) & (NB1 - 1u);
                        atomicAdd(&s_cnt[1][bin], 1u);
                        atomicAdd(&s_sum[1][bin], __uint_as_float(u));
                    }
                } else {
                    if ((u >> 10) == pref1) {
                        unsigned int bin = u & (NB1 - 1u);
                        atomicAdd(&s_cnt[1][bin], 1u);
                        atomicAdd(&s_sum[1][bin], __uint_as_float(u));
                    }
                }
            }
        }
    }
    __syncthreads();

    unsigned int (*gc)[NB1] = (level == 1) ? ws->h1_cnt : ws->h2_cnt;
    float        (*gs)[NB1] = (level == 1) ? ws->h1_sum : ws->h2_sum;
    for (unsigned int i = threadIdx.x; i < NB1; i += blockDim.x) {
        if (s_cnt[0][i])        atomicAdd(&gc[0][i], s_cnt[0][i]);
        if (s_cnt[1][i])        atomicAdd(&gc[1][i], s_cnt[1][i]);
        if (s_sum[0][i] != 0.f) atomicAdd(&gs[0][i], s_sum[0][i]);
        if (s_sum[1][i] != 0.f) atomicAdd(&gs[1][i], s_sum[1][i]);
    }
}

// Scan a histogram level from the top; level 2 also emits the final scalar.
__global__ void ohnm_select(OhnmWs* __restrict__ ws, int level,
                            float* __restrict__ d_out)
{
    __shared__ float s_loss[2];
    const int ch = (int)threadIdx.x;
    if (ch < 2) {
        if (level == 0) {
            unsigned int np = ws->num_pos[ch];
            unsigned int nn = NTOT - np;
            unsigned int k  = KMAX;
            unsigned int fp = 4u * np;
            if (fp < k) k = fp;
            if (nn < k) k = nn;
            ws->k[ch] = k;
            unsigned int cum = 0u, pbin = 0u;
            float sum = 0.f;
            for (int bb = NB0 - 1; bb >= 0; --bb) {
                unsigned int c = ws->h0_cnt[ch][bb];
                if (cum + c >= k) { pbin = (unsigned int)bb; break; }
                cum += c; sum += ws->h0_sum[ch][bb];
            }
            ws->prefix[ch] = pbin;
            ws->cnt_gt[ch] = cum;
            ws->sum_gt[ch] = sum;
            ws->k_rem[ch]  = k - cum;
        } else if (level == 1) {
            unsigned int kr = ws->k_rem[ch];
            unsigned int cum = 0u, pbin = 0u;
            float sum = 0.f;
            for (int bb = NB1 - 1; bb >= 0; --bb) {
                unsigned int c = ws->h1_cnt[ch][bb];
                if (cum + c >= kr) { pbin = (unsigned int)bb; break; }
                cum += c; sum += ws->h1_sum[ch][bb];
            }
            ws->prefix[ch] = (ws->prefix[ch] << 10) | pbin;
            ws->cnt_gt[ch] += cum;
            ws->sum_gt[ch] += sum;
            ws->k_rem[ch]   = kr - cum;
        } else {
            unsigned int kr = ws->k_rem[ch];
            unsigned int cum = 0u, pbin = 0u;
            float sum = 0.f;
            for (int bb = NB1 - 1; bb >= 0; --bb) {
                unsigned int c = ws->h2_cnt[ch][bb];
                if (cum + c >= kr) { pbin = (unsigned int)bb; break; }
                cum += c; sum += ws->h2_sum[ch][bb];
            }
            const unsigned int k      = ws->k[ch];
            const unsigned int cnt_gt = ws->cnt_gt[ch] + cum;
            const float        sum_gt = ws->sum_gt[ch] + sum;
            const float T = __uint_as_float((ws->prefix[ch] << 10) | pbin);
            float neg_sum = (k == 0u) ? 0.f
                                      : (sum_gt + (float)(k - cnt_gt) * T);
            s_loss[ch] = (ws->pos_sum[ch] + neg_sum) /
                         (float)(ws->num_pos[ch] + k);
        }
    }
    __syncthreads();
    if (level == 2 && threadIdx.x == 0u) {
        d_out[0] = s_loss[0] + s_loss[1];
    }
}

extern "C" void kernel_launch(void* const* d_in, const int* in_sizes, int n_in,
                              void* d_out, int out_size, void* d_ws, size_t ws_size,
                              hipStream_t stream) {
    const float* out_t = (const float*)d_in[0];  // (64,2,512,512)
    const float* cmap  = (const float*)d_in[1];  // (64,512,512)
    const float* amap  = (const float*)d_in[2];
    const float* cw    = (const float*)d_in[3];
    const float* aw    = (const float*)d_in[4];
    float* res = (float*)d_out;
    OhnmWs* ws = (OhnmWs*)d_ws;

    const unsigned int nwords = (unsigned int)(sizeof(OhnmWs) / 4u);
    ohnm_zero_ws<<<(nwords + 255u) / 256u, 256, 0, stream>>>((unsigned int*)d_ws, nwords);

    const int blocks = 2048;
    ohnm_pass1<<<blocks, 256, 0, stream>>>(out_t, cmap, amap, cw, aw, ws);
    ohnm_select<<<1, 32, 0, stream>>>(ws, 0, res);
    ohnm_refine<<<blocks, 256, 0, stream>>>(out_t, cmap, amap, ws, 1);
    ohnm_select<<<1, 32, 0, stream>>>(ws, 1, res);
    ohnm_refine<<<blocks, 256, 0, stream>>>(out_t, cmap, amap, ws, 2);
    ohnm_select<<<1, 32, 0, stream>>>(ws, 2, res);
}